// YOLO_54254026883511
// MI455X (gfx1250) — compile-verified
//
#include <hip/hip_runtime.h>
#include <hip/hip_bf16.h>

// ---------------------------------------------------------------------------
// YOLO decode: (32, 3*85, 64, 64) f32  ->  (32, 3*64*64, 85) f32
// Bandwidth-bound transpose + elementwise transform.
//   Phase 1: async global->LDS transposing scatter (ASYNCcnt, b32/lane)
//   Phase 2: bulk sigmoid (no index math), float4 LDS reads, b128 NT stores
//   Phase 3: fixup attrs 0..3 (box xywh) after s_wait_storecnt + barrier
// ---------------------------------------------------------------------------

#define BS    32
#define A_    3
#define H_    64
#define W_    64
#define HW    (H_ * W_)          // 4096
#define ATTRS 85
#define TILE  128                // hw positions per block
#define NELT  (TILE * ATTRS)     // 10880 floats per tile
#define N4    (NELT / 4)         // 2720 float4 per tile
#define LOG2E 1.44269504088896340736f

#if defined(__has_builtin)
#if __has_builtin(__builtin_amdgcn_global_load_async_to_lds_b32) && \
    __has_builtin(__builtin_amdgcn_s_wait_asynccnt)
#define USE_ASYNC_LDS 1
#endif
#endif
#ifndef USE_ASYNC_LDS
#define USE_ASYNC_LDS 0
#endif

typedef __attribute__((address_space(1))) int* gint_p;   // global
typedef __attribute__((address_space(3))) int* lint_p;   // LDS
typedef __attribute__((ext_vector_type(4))) float v4f;

__device__ __forceinline__ float sigmoid_fast(float x) {
    // 1 / (1 + e^-x): v_mul (neg const), v_exp, v_add, v_rcp
    return __builtin_amdgcn_rcpf(1.0f + __builtin_amdgcn_exp2f(x * (-LOG2E)));
}

__global__ __launch_bounds__(TILE) void yolo_decode_kernel(
    const float* __restrict__ in,       // (BS, A*ATTRS, H, W)
    const float* __restrict__ anchors,  // (A, 2)
    const int* __restrict__ image_h,    // scalar
    const int* __restrict__ image_w,    // scalar
    float* __restrict__ out)            // (BS, A*HW, ATTRS)
{
    __shared__ float lds[NELT];          // 43,520 B; flat idx == output idx

    const int tid  = threadIdx.x;        // 0..127
    const int tile = blockIdx.x;         // 0..31
    const int a    = blockIdx.y;         // 0..2
    const int b    = blockIdx.z;         // 0..31
    const int hw0  = tile * TILE;

    // Input: element (b, a, attr, hw) at ((b*A + a)*ATTRS + attr)*HW + hw
    const long ibase = ((long)(b * A_ + a) * ATTRS) * HW + hw0;
    // Output: element (b, a, hw, attr) at ((b*A + a)*HW + hw)*ATTRS + attr
    const long obase = ((long)(b * A_ + a) * HW + hw0) * (long)ATTRS;

    // ---- Phase 1: transposing copy global -> LDS ------------------------
    // lane `tid` owns hw = hw0 + tid; lds[tid*85 + attr].
    // LDS dword stride 85 across lanes: 85 mod 64 = 21, gcd(21,64)=1 ->
    // bank-conflict-free scatter.
#if USE_ASYNC_LDS
#pragma unroll
    for (int attr = 0; attr < ATTRS; ++attr) {
        const float* gp = in + ibase + (long)attr * HW + tid;
        __builtin_amdgcn_global_load_async_to_lds_b32(
            (gint_p)gp, (lint_p)&lds[tid * ATTRS + attr], 0, 0);
    }
    __builtin_amdgcn_s_wait_asynccnt(0);
    __syncthreads();
#else
#pragma unroll
    for (int attr = 0; attr < ATTRS; ++attr) {
        lds[tid * ATTRS + attr] = in[ibase + (long)attr * HW + tid];
    }
    __syncthreads();
#endif

    // ---- Phase 2: bulk sigmoid, vectorized ------------------------------
    // Flat LDS index == flat output index, so float4 LDS reads are 512 B
    // contiguous per wave (full 64-bank bandwidth) and stores are b128.
    // No per-element index decomposition at all.
#pragma unroll 3
    for (int i = 0; i < N4 / TILE; ++i) {            // 21 full iterations
        const int k = i * TILE + tid;                // float4 index
        const v4f x4 = *(const v4f*)&lds[4 * k];
        v4f v4;
#pragma unroll
        for (int j = 0; j < 4; ++j) v4[j] = sigmoid_fast(x4[j]);
        __builtin_nontemporal_store(v4, (v4f*)(out + obase + 4 * (long)k));
    }
    if (tid < (N4 % TILE)) {                         // tail: 32 float4
        const int k = (N4 / TILE) * TILE + tid;
        const v4f x4 = *(const v4f*)&lds[4 * k];
        v4f v4;
#pragma unroll
        for (int j = 0; j < 4; ++j) v4[j] = sigmoid_fast(x4[j]);
        __builtin_nontemporal_store(v4, (v4f*)(out + obase + 4 * (long)k));
    }

    // ---- Phase 3: fixup attrs 0..3 (box x,y,w,h) ------------------------
    // Bulk pass wrote sigmoid there; overwrite with the box transform.
    // Drain this wave's stores, then barrier, so the overwrites are ordered
    // after all bulk stores at L2.
    asm volatile("s_wait_storecnt 0x0" ::: "memory");
    __syncthreads();

    {
        const float sw = (float)(*image_w) * (1.0f / (float)W_);   // 8.0
        const float sh = (float)(*image_h) * (1.0f / (float)H_);   // 8.0
        const float aw = anchors[2 * a + 0];
        const float ah = anchors[2 * a + 1];

        const int   t  = tid;                 // this lane's hw row
        const int   hw = hw0 + t;
        const float gx = (float)(hw & (W_ - 1));
        const float gy = (float)(hw >> 6);

        // stride-85 dword reads across lanes: conflict-free
        const float x0 = lds[t * ATTRS + 0];
        const float x1 = lds[t * ATTRS + 1];
        const float x2 = lds[t * ATTRS + 2];
        const float x3 = lds[t * ATTRS + 3];

        const float bx = (sigmoid_fast(x0) + gx) * sw;
        const float by = (sigmoid_fast(x1) + gy) * sh;
        const float bw = __builtin_amdgcn_exp2f(x2 * LOG2E) * aw;
        const float bh = __builtin_amdgcn_exp2f(x3 * LOG2E) * ah;

        const long ob = obase + (long)t * ATTRS;
        __builtin_nontemporal_store(bx, out + ob + 0);
        __builtin_nontemporal_store(by, out + ob + 1);
        __builtin_nontemporal_store(bw, out + ob + 2);
        __builtin_nontemporal_store(bh, out + ob + 3);
    }
}

extern "C" void kernel_launch(void* const* d_in, const int* in_sizes, int n_in,
                              void* d_out, int out_size, void* d_ws, size_t ws_size,
                              hipStream_t stream) {
    (void)in_sizes; (void)n_in; (void)out_size; (void)d_ws; (void)ws_size;

    const float* in      = (const float*)d_in[0];
    const float* anchors = (const float*)d_in[1];
    const int*   image_h = (const int*)d_in[2];
    const int*   image_w = (const int*)d_in[3];
    float*       out     = (float*)d_out;

    dim3 grid(HW / TILE, A_, BS);   // (32, 3, 32) = 3072 blocks
    dim3 block(TILE);               // 128 threads = 4 wave32
    yolo_decode_kernel<<<grid, block, 0, stream>>>(in, anchors, image_h, image_w, out);
}